// GCN_23390391894560
// MI455X (gfx1250) — compile-verified
//
#include <hip/hip_runtime.h>
#include <hip/hip_bf16.h>

typedef __attribute__((ext_vector_type(2))) float v2f;
typedef __attribute__((ext_vector_type(8))) float v8f;

// ---------------------------------------------------------------------------
// Degree accumulation: deg_out[src[e]] += 1, deg_in[dst[e]] += 1
// ---------------------------------------------------------------------------
__global__ __launch_bounds__(256) void gcn_degree_kernel(
    const int* __restrict__ src, const int* __restrict__ dst,
    float* __restrict__ deg_out, float* __restrict__ deg_in, int E)
{
    int e = blockIdx.x * blockDim.x + threadIdx.x;
    if (e < E) {
        unsafeAtomicAdd(&deg_out[src[e]], 1.0f);
        unsafeAtomicAdd(&deg_in[dst[e]], 1.0f);
    }
}

// ---------------------------------------------------------------------------
// d[i] = (max(d[i], 1))^-1/2   (in place over both degree arrays)
// ---------------------------------------------------------------------------
__global__ __launch_bounds__(256) void gcn_invsqrt_kernel(float* __restrict__ d, int n)
{
    int i = blockIdx.x * blockDim.x + threadIdx.x;
    if (i < n) {
        float x = fmaxf(d[i], 1.0f);
        d[i] = rsqrtf(x);
    }
}

// ---------------------------------------------------------------------------
// Pure fp32 WMMA GEMM:  C[M,N] = A @ B
// Block = 256 threads = 8 waves; block tile 128(M) x 16(N); K slabs of 16.
// LDS double-buffered, next slab register-prefetched (global_load_b128) while
// 4x V_WMMA_F32_16X16X4_F32 consume the current one. The interior/edge
// decision is made ONCE per block: interior blocks run a loop with zero
// predication; only ragged blocks take the masked loop.
// A rows padded to 20 floats in LDS -> A-fragment reads are conflict-free.
// ---------------------------------------------------------------------------
__global__ __launch_bounds__(256) void gcn_wmma_gemm_f32(
    const float* __restrict__ A, const float* __restrict__ B,
    float* __restrict__ C, int M, int N, int K, int lda, int ldb, int ldc)
{
    constexpr int AP = 20;                 // padded A row stride (floats)
    __shared__ float As[2][128 * AP];      // 2 x 10 KB
    __shared__ float Bs[2][16 * 16];       // 2 x 1 KB

    const int t    = threadIdx.x;
    const int lane = t & 31;
    const int wv   = t >> 5;               // wave id 0..7 -> M sub-tile
    const int half = lane >> 4;            // 0/1: low/high 16 lanes
    const int l16  = lane & 15;

    const int m0 = blockIdx.x * 128;
    const int n0 = blockIdx.y * 16;

    // A staging: 128x16 tile = 512 float4 slots; thread owns slots t and t+256.
    const int r0 = t >> 2;                 // rows 0..63
    const int r1 = (t + 256) >> 2;         // rows 64..127
    const int c0 = (t & 3) * 4;            // col group (same for both slots)
    // B staging: one scalar per thread (16x16 tile)
    const int bk = t >> 4, bc = t & 15;

    const float* pa0 = A + (long)(m0 + r0) * lda + c0;   // advance by k0
    const float* pa1 = A + (long)(m0 + r1) * lda + c0;
    const float* pb  = B + (long)bk * ldb + n0 + bc;     // advance by k0*ldb

    float4 ra0, ra1;
    float  rb;

    v8f acc = {};

    // Shared pipeline driver; `fetchf(k0)` stages slab k0 into registers.
    auto run = [&](auto&& fetchf) {
        fetchf(0);
        int buf = 0;
        for (int k0 = 0; k0 < K; k0 += 16) {
            *(float4*)&As[buf][r0 * AP + c0] = ra0;
            *(float4*)&As[buf][r1 * AP + c0] = ra1;
            Bs[buf][bk * 16 + bc] = rb;
            __syncthreads();

            if (k0 + 16 < K) fetchf(k0 + 16);   // overlap loads with WMMA

            const float* as = &As[buf][(wv * 16 + l16) * AP];
            const float* bs = &Bs[buf][0];
            #pragma unroll
            for (int ks = 0; ks < 16; ks += 4) {
                v2f a, b;
                // A frag: lane L -> M = L%16, K = ks + 2*half + vgpr
                a.x = as[ks + half * 2 + 0];
                a.y = as[ks + half * 2 + 1];
                // B frag: lane L -> K = ks + 2*half + vgpr, N = L%16
                b.x = bs[(ks + half * 2 + 0) * 16 + l16];
                b.y = bs[(ks + half * 2 + 1) * 16 + l16];
                acc = __builtin_amdgcn_wmma_f32_16x16x4_f32(
                    false, a, false, b, (short)0, acc, false, false);
            }
            buf ^= 1;
        }
    };

    const bool interior = (m0 + 128 <= M) && (n0 + 16 <= N) && ((K & 15) == 0);
    if (interior) {
        run([&](int k0) {
            ra0 = *(const float4*)(pa0 + k0);
            ra1 = *(const float4*)(pa1 + k0);
            rb  = pb[(long)k0 * ldb];
        });
    } else {
        run([&](int k0) {
            float* p0 = (float*)&ra0;
            float* p1 = (float*)&ra1;
            const int g0 = m0 + r0, g1 = m0 + r1;
            #pragma unroll
            for (int j = 0; j < 4; ++j) {
                int gk = k0 + c0 + j;
                p0[j] = (g0 < M && gk < K) ? A[(long)g0 * lda + gk] : 0.0f;
                p1[j] = (g1 < M && gk < K) ? A[(long)g1 * lda + gk] : 0.0f;
            }
            int gk = k0 + bk, gn = n0 + bc;
            rb = (gk < K && gn < N) ? B[(long)gk * ldb + gn] : 0.0f;
        });
    }

    // store C: VGPR v -> M = v + 8*half, N = l16
    #pragma unroll
    for (int v = 0; v < 8; ++v) {
        int m = m0 + wv * 16 + v + half * 8;
        int n = n0 + l16;
        if (m < M && n < N) C[(long)m * ldc + n] = acc[v];
    }
}

// ---------------------------------------------------------------------------
// Edge scatter with fused source-side norm:
//   agg[dst[e], :] += H[src[e], :] * (ew[e] * dinv_out[src[e]])
// One wave32 per edge; float4 row gather (b128), scalar fp32 atomics into
// L2-resident agg. Requires F % 4 == 0 (200 and 20 both are).
// ---------------------------------------------------------------------------
__global__ __launch_bounds__(256) void gcn_scatter_kernel(
    const float* __restrict__ H, float* __restrict__ agg,
    const int* __restrict__ src, const int* __restrict__ dst,
    const float* __restrict__ ew, const float* __restrict__ dinv_out,
    int E, int F)
{
    int gw   = (blockIdx.x * blockDim.x + threadIdx.x) >> 5;
    int lane = threadIdx.x & 31;
    if (gw >= E) return;
    int   s   = src[gw];
    int   d   = dst[gw];
    float wgt = ew[gw] * dinv_out[s];
    const float4* hs = (const float4*)(H + (long)s * F);
    float*        ad = agg + (long)d * F;
    const int F4 = F >> 2;
    for (int f4 = lane; f4 < F4; f4 += 32) {
        float4 h = hs[f4];
        int f = f4 * 4;
        unsafeAtomicAdd(&ad[f + 0], h.x * wgt);
        unsafeAtomicAdd(&ad[f + 1], h.y * wgt);
        unsafeAtomicAdd(&ad[f + 2], h.z * wgt);
        unsafeAtomicAdd(&ad[f + 3], h.w * wgt);
    }
}

// ---------------------------------------------------------------------------
// Layer-1 epilogue (float4): out[i,j] = relu(agg[i,j] * dinv_in[i] + b[j])
// ---------------------------------------------------------------------------
__global__ __launch_bounds__(256) void gcn_epilogue_relu_kernel(
    const float* __restrict__ agg, const float* __restrict__ dinv_in,
    const float* __restrict__ b, float* __restrict__ out, int total4, int F4)
{
    int q = blockIdx.x * blockDim.x + threadIdx.x;
    if (q < total4) {
        int r = q / F4;                 // 32-bit div, amortized over 4 elems
        int c4 = q - r * F4;
        int i = q * 4;
        float s = dinv_in[r];
        float4 v  = *(const float4*)(agg + i);
        float4 bb = *(const float4*)(b + c4 * 4);
        float4 o;
        o.x = fmaxf(v.x * s + bb.x, 0.0f);
        o.y = fmaxf(v.y * s + bb.y, 0.0f);
        o.z = fmaxf(v.z * s + bb.z, 0.0f);
        o.w = fmaxf(v.w * s + bb.w, 0.0f);
        *(float4*)(out + i) = o;
    }
}

// ---------------------------------------------------------------------------
// Final epilogue (float4): out[i,j] = agg[i,j] * dinv_in[i] + b[j]
// ---------------------------------------------------------------------------
__global__ __launch_bounds__(256) void gcn_final_kernel(
    const float* __restrict__ agg, const float* __restrict__ dinv_in,
    const float* __restrict__ b, float* __restrict__ out, int total4, int F4)
{
    int q = blockIdx.x * blockDim.x + threadIdx.x;
    if (q < total4) {
        int r = q / F4;
        int c4 = q - r * F4;
        int i = q * 4;
        float s = dinv_in[r];
        float4 v  = *(const float4*)(agg + i);
        float4 bb = *(const float4*)(b + c4 * 4);
        float4 o;
        o.x = v.x * s + bb.x;
        o.y = v.y * s + bb.y;
        o.z = v.z * s + bb.z;
        o.w = v.w * s + bb.w;
        *(float4*)(out + i) = o;
    }
}

// ---------------------------------------------------------------------------
// Launcher
// ---------------------------------------------------------------------------
extern "C" void kernel_launch(void* const* d_in, const int* in_sizes, int n_in,
                              void* d_out, int out_size, void* d_ws, size_t ws_size,
                              hipStream_t stream)
{
    const float* feature = (const float*)d_in[0];
    const int*   src     = (const int*)  d_in[1];
    const int*   dst     = (const int*)  d_in[2];
    const float* edge_w  = (const float*)d_in[3];
    const float* W1      = (const float*)d_in[4];
    const float* b1      = (const float*)d_in[5];
    const float* W2      = (const float*)d_in[6];
    const float* b2      = (const float*)d_in[7];
    float*       out     = (float*)d_out;

    const int F1 = in_sizes[5];            // 200
    const int F2 = in_sizes[7];            // 20
    const int K0 = in_sizes[4] / F1;       // 768
    const int N  = in_sizes[0] / K0;       // 50000
    const int E  = in_sizes[1];            // 800000

    // Workspace layout (floats)
    float* ws       = (float*)d_ws;
    float* deg_out  = ws;                  // N  -> dinv_out in place
    float* deg_in   = ws + N;              // N  -> dinv_in in place
    float* H1       = deg_in + N;          // N * F1
    float* agg1     = H1 + (long)N * F1;   // N * F1
    float* H2       = agg1 + (long)N * F1; // N * F2
    float* agg2     = H2 + (long)N * F2;   // N * F2

    // Zero the accumulators (determinism on every call)
    hipMemsetAsync(deg_out, 0, sizeof(float) * 2 * N, stream);
    hipMemsetAsync(agg1, 0, sizeof(float) * (long)N * F1, stream);
    hipMemsetAsync(agg2, 0, sizeof(float) * (long)N * F2, stream);

    // Degrees -> deg^-1/2
    gcn_degree_kernel<<<(E + 255) / 256, 256, 0, stream>>>(src, dst, deg_out, deg_in, E);
    gcn_invsqrt_kernel<<<(2 * N + 255) / 256, 256, 0, stream>>>(deg_out, 2 * N);

    // Layer 1: H1 = X @ W1   (src-side norm folded into scatter weight)
    {
        dim3 grid((N + 127) / 128, (F1 + 15) / 16);
        gcn_wmma_gemm_f32<<<grid, 256, 0, stream>>>(feature, W1, H1, N, F1, K0, K0, F1, F1);
    }
    // agg1[dst] += H1[src] * ew * dinv_out[src]
    gcn_scatter_kernel<<<(E + 7) / 8, 256, 0, stream>>>(H1, agg1, src, dst, edge_w,
                                                        deg_out, E, F1);
    // H1 <- relu(agg1 * dinv_in + b1)   (layer-2 input, reuses H1 buffer)
    {
        int total4 = N * (F1 / 4);
        gcn_epilogue_relu_kernel<<<(total4 + 255) / 256, 256, 0, stream>>>(
            agg1, deg_in, b1, H1, total4, F1 / 4);
    }

    // Layer 2: H2 = H1 @ W2   (src-side norm folded into scatter weight)
    {
        dim3 grid((N + 127) / 128, (F2 + 15) / 16);
        gcn_wmma_gemm_f32<<<grid, 256, 0, stream>>>(H1, W2, H2, N, F2, F1, F1, F2, F2);
    }
    // agg2[dst] += H2[src] * ew * dinv_out[src]
    gcn_scatter_kernel<<<(E + 7) / 8, 256, 0, stream>>>(H2, agg2, src, dst, edge_w,
                                                        deg_out, E, F2);
    // out = agg2 * dinv_in + b2
    {
        int total4 = N * (F2 / 4);
        gcn_final_kernel<<<(total4 + 255) / 256, 256, 0, stream>>>(
            agg2, deg_in, b2, out, total4, F2 / 4);
    }
}